// DevConv_74586402063285
// MI455X (gfx1250) — compile-verified
//
#include <hip/hip_runtime.h>
#include <math.h>

#define N_NODES 2048
#define DIM 32
#define JT 512                       // j-tile size (rows of p per LDS buffer)
#define NTILES (N_NODES / JT)        // 4
#define ROWS_PER_BLOCK 8             // one wave per row
#define TILE_FLOATS (JT * DIM)       // 16384 floats = 64 KB per buffer

typedef __attribute__((ext_vector_type(2))) float v2f;
typedef __attribute__((ext_vector_type(8))) float v8f;
typedef __attribute__((ext_vector_type(4))) int   v4i;

// Address-space-qualified v4i pointers for the async-to-LDS builtin
// (clang prints AS1 as `__device__` in HIP mode; param0 is `v4i AS1*`).
typedef __attribute__((address_space(1))) v4i as1_v4i;
typedef __attribute__((address_space(3))) v4i as3_v4i;

#if __has_builtin(__builtin_amdgcn_global_load_async_to_lds_b128)
#define HAVE_ASYNC 1
#else
#define HAVE_ASYNC 0
#endif

// -----------------------------------------------------------------------------
// Kernel 1: p[j][d] = sum_k nodes[j][k] * W_theta[k][d]
// One wave per 16(M) x 32(D) tile via two V_WMMA_F32_16X16X4_F32 (K 3->4 pad).
// (Round-1 verified: lowers to v_wmma_f32_16x16x4_f32.)
// -----------------------------------------------------------------------------
__global__ __launch_bounds__(32) void proj_wmma(const float* __restrict__ nodes,  // [N][3]
                                                const float* __restrict__ Wth,    // [3][DIM]
                                                float* __restrict__ p)            // [N][DIM]
{
    const int lane  = threadIdx.x & 31;
    const int half  = lane >> 4;   // 0: lanes 0-15 (K=0,1)  1: lanes 16-31 (K=2,3)
    const int lr    = lane & 15;
    const int mBase = blockIdx.x * 16;

    const int m = mBase + lr;
    v2f a;
    a.x = half ? nodes[m * 3 + 2] : nodes[m * 3 + 0];
    a.y = half ? 0.0f             : nodes[m * 3 + 1];

#pragma unroll
    for (int nt = 0; nt < 2; ++nt) {
        const int n = nt * 16 + lr;
        v2f b;
        b.x = half ? Wth[2 * DIM + n] : Wth[0 * DIM + n];
        b.y = half ? 0.0f             : Wth[1 * DIM + n];

        v8f c = {};
        c = __builtin_amdgcn_wmma_f32_16x16x4_f32(false, a, false, b,
                                                  (short)0, c, false, false);
        const int mrow = mBase + (half ? 8 : 0);
#pragma unroll
        for (int r = 0; r < 8; ++r)
            p[(mrow + r) * DIM + nt * 16 + lr] = c[r];
    }
}

// -----------------------------------------------------------------------------
// Async fill of one 64 KB p-tile into LDS (256 threads, 16 x b128 each).
// -----------------------------------------------------------------------------
static __device__ inline void issue_tile_fill(float* __restrict__ dst_lds,
                                              const float* __restrict__ src_glb,
                                              int tid) {
#if HAVE_ASYNC
#pragma unroll
    for (int k = 0; k < TILE_FLOATS / 4 / 256; ++k) {
        const int idx = (k * 256 + tid) * 4;
        __builtin_amdgcn_global_load_async_to_lds_b128(
            (as1_v4i*)(src_glb + idx),
            (as3_v4i*)(dst_lds + idx),
            0, 0);
    }
#else
#pragma unroll
    for (int k = 0; k < TILE_FLOATS / 4 / 256; ++k) {
        const int idx = k * 256 + tid;
        ((float4*)dst_lds)[idx] = ((const float4*)src_glb)[idx];
    }
#endif
}

static __device__ inline void wait_async_done() {
#if HAVE_ASYNC
#if __has_builtin(__builtin_amdgcn_s_wait_asynccnt)
    __builtin_amdgcn_s_wait_asynccnt(0);
#else
    asm volatile("s_wait_asynccnt 0x0" ::: "memory");
#endif
#endif
}

// -----------------------------------------------------------------------------
// Kernel 2: streaming masked row-max + weighted mean + residual.
// Block = 8 waves = 8 rows; wave lane = feature d. p tiles live in LDS
// (double-buffered, async-filled); adjacency is the single 16 MB HBM stream,
// read once as b128. The per-(i,j) guard is a wave-uniform SGPR bit, so the
// inner body is: pipelined ds_load + (scalar-guarded) v_max.
// -----------------------------------------------------------------------------
__global__ __launch_bounds__(256) void masked_max_kernel(
    const float* __restrict__ prev,   // [N]
    const int*   __restrict__ adj,    // [N][N]
    const float* __restrict__ Wphi,   // [DIM]
    const float* __restrict__ p,      // [N][DIM]
    float*       __restrict__ out)    // [N]
{
    __shared__ float sp[2][TILE_FLOATS];   // 128 KB: double-buffered p tiles

    const int tid  = threadIdx.x;
    const int lane = tid & 31;
    const int wave = tid >> 5;
    const int i    = blockIdx.x * ROWS_PER_BLOCK + wave;

    float m  = -INFINITY;   // running max over adjacent j of p[j][lane]
    int   nz = 0;           // number of adjacent neighbors in row i

    issue_tile_fill(sp[0], p, tid);   // tile 0

    for (int jt = 0; jt < NTILES; ++jt) {
        const int cur = jt & 1;
        wait_async_done();
        __syncthreads();              // tile `cur` ready; prior reads of cur^1 done
        if (jt + 1 < NTILES)
            issue_tile_fill(sp[cur ^ 1], p + (size_t)(jt + 1) * TILE_FLOATS, tid);

        const int*   aseg = adj + (size_t)i * N_NODES + jt * JT;
        const float* spc  = sp[cur];

        for (int jb = 0; jb < JT; jb += 128) {
            // j = jt*JT + jb + 4*lane + q  -> bit l of bq[q] is j = jb + 4*l + q
            const int4 a4 = ((const int4*)(aseg + jb))[lane];
            unsigned bq0 = (unsigned)__ballot(a4.x != 0);
            unsigned bq1 = (unsigned)__ballot(a4.y != 0);
            unsigned bq2 = (unsigned)__ballot(a4.z != 0);
            unsigned bq3 = (unsigned)__ballot(a4.w != 0);
            nz += __popc(bq0) + __popc(bq1) + __popc(bq2) + __popc(bq3);

#pragma unroll
            for (int l = 0; l < 32; ++l) {
                // Unconditional LDS loads (pipelined), scalar-guarded maxes.
                float pv0 = spc[(jb + 4 * l + 0) * DIM + lane];
                float pv1 = spc[(jb + 4 * l + 1) * DIM + lane];
                float pv2 = spc[(jb + 4 * l + 2) * DIM + lane];
                float pv3 = spc[(jb + 4 * l + 3) * DIM + lane];
                if ((bq0 >> l) & 1) m = fmaxf(m, pv0);
                if ((bq1 >> l) & 1) m = fmaxf(m, pv1);
                if ((bq2 >> l) & 1) m = fmaxf(m, pv2);
                if ((bq3 >> l) & 1) m = fmaxf(m, pv3);
            }
        }
        __syncthreads();              // all waves done reading tile `cur`
    }

    // max_j adj*(p_j - p_i) == (max_{adj} p_j) - p_i, clamped by the zeros the
    // mask injects whenever row i is not all-ones; m == -inf (no neighbor)
    // collapses to 0 via the clamp.
    float v = m - p[(size_t)i * DIM + lane];
    if (nz < N_NODES) v = fmaxf(v, 0.0f);

    float contrib = Wphi[lane] * v;
#pragma unroll
    for (int off = 16; off >= 1; off >>= 1)
        contrib += __shfl_xor(contrib, off, 32);

    if (lane == 0) out[i] = prev[i] + contrib * (1.0f / DIM);
}

// -----------------------------------------------------------------------------
// Launch. Inputs (setup_inputs order):
//   0: previous_inclusion_score [N]      f32
//   1: nodes                    [N][3]   f32
//   2: adjacency_matrix         [N][N]   i32
//   3: W_phi                    [DIM]    f32
//   4: W_theta                  [3][DIM] f32
// d_ws: p scratch, N*DIM*4 = 256 KB.
// -----------------------------------------------------------------------------
extern "C" void kernel_launch(void* const* d_in, const int* in_sizes, int n_in,
                              void* d_out, int out_size, void* d_ws, size_t ws_size,
                              hipStream_t stream) {
    const float* prev  = (const float*)d_in[0];
    const float* nodes = (const float*)d_in[1];
    const int*   adj   = (const int*)  d_in[2];
    const float* Wphi  = (const float*)d_in[3];
    const float* Wth   = (const float*)d_in[4];
    float* out = (float*)d_out;
    float* p   = (float*)d_ws;   // [N][DIM]

    proj_wmma<<<N_NODES / 16, 32, 0, stream>>>(nodes, Wth, p);
    masked_max_kernel<<<N_NODES / ROWS_PER_BLOCK, 256, 0, stream>>>(prev, adj, Wphi, p, out);
}